// GCN_32212254720504
// MI455X (gfx1250) — compile-verified
//
#include <hip/hip_runtime.h>
#include <hip/hip_bf16.h>
#include <math.h>
#include <stdint.h>

// ---------------------------------------------------------------------------
// GCN (2x GCNConv + linear + log_softmax) for MI455X (gfx1250, wave32, WMMA)
//   N = 100000 nodes, E = 1.6M edges, dims 128 -> 64 -> 64 -> 40
// Dense transforms: V_WMMA_F32_16X16X4_F32 (fp32 in/out, full precision).
// A tiles staged LDS via Tensor Data Mover (tensor_load_to_lds, TENSORcnt),
// with TDM hardware row padding (K+1 dword stride) for bank-conflict-free
// fragment reads. B staged cooperatively (tiny).
// ---------------------------------------------------------------------------

typedef __attribute__((ext_vector_type(2))) float v2f;
typedef __attribute__((ext_vector_type(8))) float v8f;
typedef __attribute__((ext_vector_type(4))) unsigned int u32x4;
typedef __attribute__((ext_vector_type(8))) int i32x8;
typedef __attribute__((ext_vector_type(4))) int i32x4;

#define NN 100000
#define DHID 64

// ---------------- degree / norm ----------------

__global__ __launch_bounds__(256) void init_deg_kernel(float* __restrict__ deg, int n) {
    int i = blockIdx.x * 256 + threadIdx.x;
    if (i < n) deg[i] = 1.0f;  // self-loop contributes 1 to every node's degree
}

__global__ __launch_bounds__(256) void edge_deg_kernel(const int* __restrict__ dst,
                                                       float* __restrict__ deg, int E) {
    int e = blockIdx.x * 256 + threadIdx.x;
    if (e < E) atomicAdd(&deg[dst[e]], 1.0f);
}

__global__ __launch_bounds__(256) void dinv_kernel(const float* __restrict__ deg,
                                                   float* __restrict__ dinv, int n) {
    int i = blockIdx.x * 256 + threadIdx.x;
    if (i < n) {
        float d = deg[i];
        dinv[i] = (d > 0.0f) ? rsqrtf(d) : 0.0f;
    }
}

// ---------------- WMMA GEMM: C[M x realN] = A[M x K] @ B[K x realN] ----------------
// Block = 256 threads = 8 waves, 128 rows. A tile DMA'd into LDS by the TDM
// (async tensor load, 1-dword row pad). Each wave computes one 16-row strip
// across NTILES 16-col tiles with f32 WMMA. LAST adds bias + masks cols.

template <int K, int NTILES, bool LAST>
__global__ __launch_bounds__(256) void gemm_wmma_kernel(const float* __restrict__ A,
                                                        const float* __restrict__ Bw,
                                                        const float* __restrict__ bias,
                                                        float* __restrict__ C,
                                                        int M, int realN) {
    constexpr int N = NTILES * 16;
    constexpr int ASTRIDE = K + 1;  // padded row stride in dwords (TDM pad_amount=1)
    __shared__ float sB[K * N];
    __shared__ float sA[128 * ASTRIDE];

    // stage B (K x realN), zero-padded to N cols
    for (int i = threadIdx.x; i < K * N; i += 256) {
        int kk = i / N, nn = i % N;
        sB[i] = (nn < realN) ? Bw[kk * realN + nn] : 0.0f;
    }

    const int row0 = blockIdx.x * 128;

    // --- TDM: async DMA of A[row0 : row0+128, 0:K] into sA (row stride K+1) ---
    if (threadIdx.x < 32) {  // wave 0 issues once (TDM ignores EXEC)
        uint64_t gaddr = (uint64_t)(uintptr_t)(A + (size_t)row0 * K);
        uint32_t ldsoff = (uint32_t)(uintptr_t)(&sA[0]);
        uint32_t remRows = (uint32_t)(M - row0);  // OOB rows read back as zero

        u32x4 g0;
        g0[0] = 1u;                                              // count=1, user mode
        g0[1] = ldsoff;                                          // lds_addr (bytes)
        g0[2] = (uint32_t)(gaddr & 0xFFFFFFFFu);                 // global_addr lo
        g0[3] = (uint32_t)((gaddr >> 32) & 0x01FFFFFFu) | (2u << 30);  // addr hi | type=2

        constexpr uint32_t padIv = (K == 128) ? 6u : 5u;  // pad every 128 / 64 dwords
        i32x8 g1;
        g1[0] = (int)((2u << 16)      // data_size = 4 bytes
                      | (1u << 20)    // pad_enable
                      | (padIv << 22) // pad_interval
                      | (0u << 25));  // pad_amount = 1 dword
        g1[1] = (int)(((uint32_t)K & 0xFFFFu) << 16);            // tensor_dim0[15:0]
        g1[2] = (int)((remRows & 0xFFFFu) << 16);                // dim0 hi=0 | tensor_dim1 lo
        g1[3] = (int)(((remRows >> 16) & 0xFFFFu)                // tensor_dim1 hi
                      | ((uint32_t)K << 16));                    // tile_dim0 = K
        g1[4] = 128;                                             // tile_dim1=128 | tile_dim2=0
        g1[5] = K;                                               // tensor_dim0_stride lo
        g1[6] = 0;
        g1[7] = 0;

        i32x4 gz = {0, 0, 0, 0};
#if __clang_major__ >= 23
        i32x8 gz8 = {0, 0, 0, 0, 0, 0, 0, 0};
        __builtin_amdgcn_tensor_load_to_lds(g0, g1, gz, gz, gz8, 0);
#else
        __builtin_amdgcn_tensor_load_to_lds(g0, g1, gz, gz, 0);
#endif
        __builtin_amdgcn_s_wait_tensorcnt(0);
    }
    __syncthreads();

    const int wave = threadIdx.x >> 5;  // 0..7
    const int lane = threadIdx.x & 31;
    const int m0 = row0 + wave * 16;
    if (m0 >= M) return;  // wave-uniform: EXEC stays all-1 for WMMA

    const int half = lane >> 4;  // 0: lanes 0-15, 1: lanes 16-31
    const int l16 = lane & 15;

    v8f acc[NTILES];
#pragma unroll
    for (int t = 0; t < NTILES; t++)
#pragma unroll
        for (int j = 0; j < 8; j++) acc[t][j] = 0.0f;

    // A-fragment (ISA 7.12.2, 32-bit A 16x4): half 0 -> K = k,k+1 ; half 1 -> K = k+2,k+3
    const float* arow = &sA[(wave * 16 + l16) * ASTRIDE];

    for (int k = 0; k < K; k += 4) {
        v2f a;
        a.x = arow[k + 2 * half + 0];
        a.y = arow[k + 2 * half + 1];
#pragma unroll
        for (int t = 0; t < NTILES; t++) {
            const int n0 = t * 16;
            v2f b;  // B 4x16 row-striped: VGPR0 = rows {k, k+2}, VGPR1 = rows {k+1, k+3}
            b.x = sB[(k + 2 * half + 0) * N + n0 + l16];
            b.y = sB[(k + 2 * half + 1) * N + n0 + l16];
            acc[t] = __builtin_amdgcn_wmma_f32_16x16x4_f32(
                false, a, false, b, (short)0, acc[t], false, false);
        }
    }

    // C/D layout: VGPR r -> row m0+r (lanes 0-15) / row m0+8+r (lanes 16-31)
#pragma unroll
    for (int t = 0; t < NTILES; t++) {
        const int col = t * 16 + l16;
#pragma unroll
        for (int r = 0; r < 8; r++) {
            const int row = m0 + r + 8 * half;
            float v = acc[t][r];
            if (LAST) {
                if (col < realN) C[(size_t)row * realN + col] = v + bias[col];
            } else {
                C[(size_t)row * realN + col] = v;
            }
        }
    }
}

// ---------------- edge scatter: agg[dst] += h[src] * dinv[src]*dinv[dst] ----------------
// 16 lanes per edge, float4 per lane (64 dims), global fp32 atomics (L2-resident).

__global__ __launch_bounds__(256) void scatter_kernel(const float* __restrict__ h,
                                                      const int* __restrict__ src,
                                                      const int* __restrict__ dst,
                                                      const float* __restrict__ dinv,
                                                      float* __restrict__ agg, int E) {
    int t = blockIdx.x * 256 + threadIdx.x;
    int e = t >> 4;
    if (e >= E) return;
    int sub = t & 15;
    int s = src[e];
    int d = dst[e];
    float nrm = dinv[s] * dinv[d];
    float4 v = reinterpret_cast<const float4*>(h + (size_t)s * DHID)[sub];
    float* op = agg + (size_t)d * DHID + sub * 4;
    atomicAdd(op + 0, v.x * nrm);
    atomicAdd(op + 1, v.y * nrm);
    atomicAdd(op + 2, v.z * nrm);
    atomicAdd(op + 3, v.w * nrm);
}

// ---------------- finalize: agg = [relu](agg + h*dinv^2 + bias), in-place ----------------

template <bool RELU>
__global__ __launch_bounds__(256) void finalize_kernel(float* __restrict__ agg,
                                                       const float* __restrict__ h,
                                                       const float* __restrict__ dinv,
                                                       const float* __restrict__ bias,
                                                       int n_nodes) {
    int idx = blockIdx.x * 256 + threadIdx.x;
    int total = n_nodes * DHID;
    if (idx >= total) return;
    int node = idx >> 6;  // /64
    int dd = idx & 63;
    float di = dinv[node];
    float v = agg[idx] + h[idx] * di * di + bias[dd];
    if (RELU) v = fmaxf(v, 0.0f);
    agg[idx] = v;
}

// ---------------- log_softmax over 40 logits per row ----------------

__global__ __launch_bounds__(256) void logsoftmax_kernel(const float* __restrict__ logits,
                                                         float* __restrict__ out, int n_rows) {
    int r = blockIdx.x * 256 + threadIdx.x;
    if (r >= n_rows) return;
    const float* p = logits + (size_t)r * 40;
    float m = -INFINITY;
#pragma unroll
    for (int j = 0; j < 40; j++) m = fmaxf(m, p[j]);
    float s = 0.0f;
#pragma unroll
    for (int j = 0; j < 40; j++) s += expf(p[j] - m);
    float ls = logf(s);
    float* o = out + (size_t)r * 40;
#pragma unroll
    for (int j = 0; j < 40; j++) o[j] = p[j] - m - ls;
}

// ---------------- host orchestration ----------------

extern "C" void kernel_launch(void* const* d_in, const int* in_sizes, int n_in,
                              void* d_out, int out_size, void* d_ws, size_t ws_size,
                              hipStream_t stream) {
    const float* x  = (const float*)d_in[0];
    const int*   ei = (const int*)d_in[1];   // [2, E] flat: row 0 = src, row 1 = dst
    const float* W1 = (const float*)d_in[2];
    const float* b1 = (const float*)d_in[3];
    const float* W2 = (const float*)d_in[4];
    const float* b2 = (const float*)d_in[5];
    const float* Wl = (const float*)d_in[6];
    const float* bl = (const float*)d_in[7];
    float* out = (float*)d_out;

    const int N = NN;
    const int E = in_sizes[1] / 2;
    const int* src = ei;
    const int* dst = ei + E;

    auto alignup = [](size_t v) { return (v + 255) & ~(size_t)255; };
    char* ws = (char*)d_ws;
    const size_t nodeVecBytes = (size_t)N * DHID * sizeof(float);
    float* deg  = (float*)ws;  ws += alignup((size_t)N * sizeof(float));
    float* dinv = (float*)ws;  ws += alignup((size_t)N * sizeof(float));
    float* buf0 = (float*)ws;  ws += alignup(nodeVecBytes);  // h1, later agg2/emb
    float* buf1 = (float*)ws;  ws += alignup(nodeVecBytes);  // agg1/h1relu, later logits
    float* buf2 = (float*)ws;  ws += alignup(nodeVecBytes);  // h2

    const int nodeBlocks = (N + 255) / 256;
    const int edgeBlocks = (E + 255) / 256;
    const int elemBlocks = (N * DHID + 255) / 256;
    const int scatBlocks = (E * 16 + 255) / 256;
    const int gemmBlocks = (N + 127) / 128;  // 782, 128 rows per block

    // degree + symmetric norm
    init_deg_kernel<<<nodeBlocks, 256, 0, stream>>>(deg, N);
    edge_deg_kernel<<<edgeBlocks, 256, 0, stream>>>(dst, deg, E);
    dinv_kernel<<<nodeBlocks, 256, 0, stream>>>(deg, dinv, N);

    // layer 1: h1 = x @ W1 ; aggregate ; relu(agg + h1*dinv^2 + b1)
    gemm_wmma_kernel<128, 4, false><<<gemmBlocks, 256, 0, stream>>>(x, W1, nullptr, buf0, N, DHID);
    hipMemsetAsync(buf1, 0, nodeVecBytes, stream);
    scatter_kernel<<<scatBlocks, 256, 0, stream>>>(buf0, src, dst, dinv, buf1, E);
    finalize_kernel<true><<<elemBlocks, 256, 0, stream>>>(buf1, buf0, dinv, b1, N);

    // layer 2: h2 = relu1 @ W2 ; aggregate ; agg + h2*dinv^2 + b2
    gemm_wmma_kernel<64, 4, false><<<gemmBlocks, 256, 0, stream>>>(buf1, W2, nullptr, buf2, N, DHID);
    hipMemsetAsync(buf0, 0, nodeVecBytes, stream);
    scatter_kernel<<<scatBlocks, 256, 0, stream>>>(buf2, src, dst, dinv, buf0, E);
    finalize_kernel<false><<<elemBlocks, 256, 0, stream>>>(buf0, buf2, dinv, b2, N);

    // head: logits = emb @ Wl + bl (N padded 40->48 in LDS) ; log_softmax
    gemm_wmma_kernel<64, 3, true><<<gemmBlocks, 256, 0, stream>>>(buf0, Wl, bl, buf1, N, 40);
    logsoftmax_kernel<<<nodeBlocks, 256, 0, stream>>>(buf1, out, N);
}